// PerturbedTopK_88407606821496
// MI455X (gfx1250) — compile-verified
//
#include <hip/hip_runtime.h>
#include <stdint.h>

#define B_DIM   32
#define D_DIM   2048
#define NSAMP   500
#define TOPK    128
#define SIGMA   0.05f
#define TPB     256
#define NWAVES  (TPB / 32)
#define EPT     (D_DIM / TPB)   // 8 elements per thread

// ---------------------------------------------------------------------------
// Kernel 1: zero the indicators region (atomics accumulate into it each call).
// ---------------------------------------------------------------------------
__global__ __launch_bounds__(TPB) void zero_kernel(float* __restrict__ p, int n4) {
    int i = blockIdx.x * TPB + threadIdx.x;
    if (i < n4) {
        ((float4*)p)[i] = make_float4(0.f, 0.f, 0.f, 0.f);
    }
}

// ---------------------------------------------------------------------------
// Block-wide inclusive scan over 256 lanes (8 wave32's):
//   intra-wave via __shfl_up (ds_bpermute lane-permute path, no barriers),
//   cross-wave via 8-word LDS broadcast. 2 barriers total.
// Returns inclusive scan; *totalOut = sum over all 256 inputs.
// ---------------------------------------------------------------------------
__device__ __forceinline__ uint32_t block_scan(uint32_t v, int tid,
                                               uint32_t* s_wsum,
                                               uint32_t* totalOut) {
    const int lane = tid & 31;
    const int wave = tid >> 5;
    uint32_t x = v;
#pragma unroll
    for (int s = 1; s < 32; s <<= 1) {
        uint32_t t = __shfl_up(x, s, 32);
        if (lane >= s) x += t;
    }
    if (lane == 31) s_wsum[wave] = x;
    __syncthreads();
    uint32_t off = 0, tot = 0;
#pragma unroll
    for (int w = 0; w < NWAVES; ++w) {
        const uint32_t ws = s_wsum[w];      // broadcast read, conflict-free
        tot += ws;
        if (w < wave) off += ws;
    }
    __syncthreads();                        // s_wsum safely reusable after this
    *totalOut = tot;
    return x + off;
}

// ---------------------------------------------------------------------------
// Kernel 2: one workgroup per (b, n) row.
//   - async-copy noise row (8KB) into LDS (global_load_async_to_lds_b128)
//   - build orderable keys in registers
//   - 4-pass MSB radix select via LDS histograms (ds_add) + hierarchical scans
//   - rank selected indices ascending, write idx + atomic scatter indicators
// ---------------------------------------------------------------------------
__global__ __launch_bounds__(TPB) void ptopk_kernel(const float* __restrict__ x,
                                                    const float* __restrict__ noise,
                                                    float* __restrict__ indicators,
                                                    float* __restrict__ idx_out) {
    __shared__ float    s_noise[D_DIM];     // 8 KB staging, written by async DMA
    __shared__ uint32_t s_hist[256];        // radix histogram
    __shared__ uint32_t s_wsum[NWAVES];     // per-wave scan partials
    __shared__ uint32_t s_bPrefix;
    __shared__ int      s_bWant;

    const int tid = threadIdx.x;
    const int wg  = blockIdx.x;             // 0 .. B*NSAMP-1
    const int b   = wg / NSAMP;

    // ---- CDNA5 async global->LDS copy of this row's noise (ASYNCcnt path) ----
    {
        const uint64_t rowBase = (uint64_t)(uintptr_t)(noise + (size_t)wg * D_DIM);
        const uint32_t ldsBase = (uint32_t)(uintptr_t)(&s_noise[0]);
#pragma unroll
        for (int it = 0; it < (D_DIM * 4) / (TPB * 16); ++it) {
            const int      chunk   = it * TPB + tid;       // 16-byte chunks
            const uint32_t ldsAddr = ldsBase + (uint32_t)chunk * 16u;
            const uint64_t gAddr   = rowBase + (uint64_t)chunk * 16u;
            asm volatile("global_load_async_to_lds_b128 %0, %1, off"
                         :: "v"(ldsAddr), "v"(gAddr) : "memory");
        }
        asm volatile("s_wait_asynccnt 0" ::: "memory");
    }
    __syncthreads();

    // ---- build monotone-orderable keys in registers (ascending d per thread) ----
    uint32_t key[EPT];
#pragma unroll
    for (int e = 0; e < EPT; ++e) {
        const int   d = tid * EPT + e;
        const float v = x[b * D_DIM + d] + SIGMA * s_noise[d];
        uint32_t    u = __float_as_uint(v);
        key[e] = (u & 0x80000000u) ? ~u : (u | 0x80000000u);  // bigger float -> bigger uint
    }

    // ---- MSB-first radix select of the K-th largest key ----
    uint32_t prefix = 0, mask = 0;
    int want = TOPK;
#pragma unroll
    for (int pass = 0; pass < 4; ++pass) {
        const int shift = 24 - pass * 8;

        s_hist[tid] = 0u;
        __syncthreads();
#pragma unroll
        for (int e = 0; e < EPT; ++e) {
            if ((key[e] & mask) == prefix)
                atomicAdd(&s_hist[(key[e] >> shift) & 255u], 1u);   // ds_add_u32
        }
        __syncthreads();

        // suffix[d] = total - exclusive[d]; derived from one forward scan
        const uint32_t cnt = s_hist[tid];
        uint32_t total;
        const uint32_t incl  = block_scan(cnt, tid, s_wsum, &total);
        const uint32_t suffD  = total - (incl - cnt);   // count with digit >= tid
        const uint32_t suffD1 = total - incl;           // count with digit >  tid
        if (suffD >= (uint32_t)want && suffD1 < (uint32_t)want) {   // unique digit
            s_bPrefix = prefix | ((uint32_t)tid << shift);
            s_bWant   = want - (int)suffD1;
        }
        __syncthreads();
        prefix = s_bPrefix;
        want   = s_bWant;
        mask  |= (0xFFu << shift);
    }

    const uint32_t T     = prefix;   // K-th largest key value
    const int      needE = want;     // # elems equal to T to include (lowest d first)

    // ---- cap ties: exclusive scan of per-thread equal-counts ----
    uint32_t cntE = 0;
#pragma unroll
    for (int e = 0; e < EPT; ++e) cntE += (key[e] == T) ? 1u : 0u;
    uint32_t totE;
    const uint32_t baseE = block_scan(cntE, tid, s_wsum, &totE) - cntE;

    // ---- selection flags (greater-than, or equal within tie budget) ----
    uint32_t selMask = 0, eq = 0;
#pragma unroll
    for (int e = 0; e < EPT; ++e) {
        bool sel = key[e] > T;
        if (key[e] == T) { sel = (baseE + eq) < (uint32_t)needE; ++eq; }
        selMask |= (sel ? 1u : 0u) << e;
    }
    const uint32_t cntS = (uint32_t)__popc(selMask);

    // ---- ascending-index ranks: exclusive scan of selected counts ----
    uint32_t totS;
    uint32_t rank = block_scan(cntS, tid, s_wsum, &totS) - cntS;

    // ---- emit sorted indices + scatter 1/N into indicators[b, rank, d] ----
    const float invN = 1.0f / (float)NSAMP;
#pragma unroll
    for (int e = 0; e < EPT; ++e) {
        if (selMask & (1u << e)) {
            const int d = tid * EPT + e;
            idx_out[(size_t)wg * TOPK + rank] = (float)d;
            atomicAdd(&indicators[((size_t)b * TOPK + rank) * D_DIM + d], invN);
            ++rank;
        }
    }
}

// ---------------------------------------------------------------------------
extern "C" void kernel_launch(void* const* d_in, const int* in_sizes, int n_in,
                              void* d_out, int out_size, void* d_ws, size_t ws_size,
                              hipStream_t stream) {
    (void)in_sizes; (void)n_in; (void)out_size; (void)d_ws; (void)ws_size;

    const float* x     = (const float*)d_in[0];   // (32, 2048)
    const float* noise = (const float*)d_in[1];   // (32, 500, 2048)

    float* out        = (float*)d_out;
    float* indicators = out;                                   // (32, 128, 2048)
    float* idx_out    = out + (size_t)B_DIM * TOPK * D_DIM;    // (32, 500, 128) as float

    const int n4 = (B_DIM * TOPK * D_DIM) / 4;
    zero_kernel<<<(n4 + TPB - 1) / TPB, TPB, 0, stream>>>(indicators, n4);

    ptopk_kernel<<<B_DIM * NSAMP, TPB, 0, stream>>>(x, noise, indicators, idx_out);
}